// DynamicKGating_4681514352968
// MI455X (gfx1250) — compile-verified
//
#include <hip/hip_runtime.h>

typedef float v2f __attribute__((ext_vector_type(2)));
typedef float v4f __attribute__((ext_vector_type(4)));
typedef float v8f __attribute__((ext_vector_type(8)));

#define BB 8
#define NN 2048
#define DD 1024
#define EE 64
#define TOPK 8
#define CAP 40
#define TAUV 0.7f
#define NTOK (BB*NN)              /* 16384 tokens */
#define NSLOT (NTOK*TOPK)         /* 131072 assignment slots */
#define DSZ ((size_t)NTOK*EE*CAP) /* 41943040 floats per output tensor */

// One K=4 GEMM micro-step: A frag (fp32 16x4) + 4 B frags (4x16) + 4 WMMAs.
#define GEMM_STEP(K0)                                                          \
    do {                                                                       \
        const int kb = (K0) + (half << 1);                                     \
        v2f a = *(const v2f*)(xrow + kb);                                      \
        const float* w0 = w + (size_t)kb * EE;                                 \
        v2f b0 = { w0[ 0 + l15], w0[EE +  0 + l15] };                          \
        v2f b1 = { w0[16 + l15], w0[EE + 16 + l15] };                          \
        v2f b2 = { w0[32 + l15], w0[EE + 32 + l15] };                          \
        v2f b3 = { w0[48 + l15], w0[EE + 48 + l15] };                          \
        acc0 = __builtin_amdgcn_wmma_f32_16x16x4_f32(false, a, false, b0,      \
                                                     (short)0, acc0, false, false); \
        acc1 = __builtin_amdgcn_wmma_f32_16x16x4_f32(false, a, false, b1,      \
                                                     (short)0, acc1, false, false); \
        acc2 = __builtin_amdgcn_wmma_f32_16x16x4_f32(false, a, false, b2,      \
                                                     (short)0, acc2, false, false); \
        acc3 = __builtin_amdgcn_wmma_f32_16x16x4_f32(false, a, false, b3,      \
                                                     (short)0, acc3, false, false); \
    } while (0)

// ---------------------------------------------------------------------------
// Kernel 1: router. One wave (32 threads) computes a 16-token x 64-expert
// logits tile with fp32 WMMA (16x16x4), then softmax + dynamic top-k.
// ---------------------------------------------------------------------------
__global__ __launch_bounds__(32)
void router_kernel(const float* __restrict__ x, const float* __restrict__ w,
                   int* __restrict__ sel_idx, float* __restrict__ sel_p)
{
    __shared__ float lds[16 * 65];          // 16 rows x 64 logits, stride 65
    const int lane = threadIdx.x;
    const int half = lane >> 4;             // 0 -> lanes 0-15, 1 -> lanes 16-31
    const int l15  = lane & 15;
    const int t0   = blockIdx.x * 16;       // first token of this tile

    v8f acc0 = {}, acc1 = {}, acc2 = {}, acc3 = {};

    // A layout (32-bit 16x4): v0: K=k0 (lanes0-15) / K=k0+2 (lanes16-31); v1: +1
    // B layout (4x16):        v0: K=k0 (lanes0-15) / K=k0+2 (lanes16-31); v1: +1
    const int arow = t0 + l15;
    const float* xrow = x + (size_t)arow * DD;

    int k0 = 0;
    // main loop: unconditionally prefetch x one K-super-block ahead (WGP scope)
    for (; k0 < DD - 64; k0 += 4) {
        __builtin_prefetch(xrow + k0 + (half << 1) + 64, 0, 3);
        GEMM_STEP(k0);
    }
    // tail: last 64 K-values, no prefetch (keeps addresses in-bounds)
    for (; k0 < DD; k0 += 4) {
        GEMM_STEP(k0);
    }

    // C/D layout: VGPR j -> row M=j (lanes0-15), M=j+8 (lanes16-31), col=lane&15
    #pragma unroll
    for (int j = 0; j < 8; ++j) {
        const int rr = j + (half << 3);
        lds[rr * 65 +  0 + l15] = acc0[j];
        lds[rr * 65 + 16 + l15] = acc1[j];
        lds[rr * 65 + 32 + l15] = acc2[j];
        lds[rr * 65 + 48 + l15] = acc3[j];
    }
    __syncthreads();

    // One lane per token row: softmax, top-8, dynamic-k cutoff, renormalize.
    if (lane < 16) {
        float* row = lds + lane * 65;
        float mx = row[0];
        for (int i = 1; i < EE; ++i) mx = fmaxf(mx, row[i]);
        float sum = 0.f;
        for (int i = 0; i < EE; ++i) { float ev = __expf(row[i] - mx); row[i] = ev; sum += ev; }
        const float inv = 1.0f / sum;

        float ptop[TOPK]; int itop[TOPK];
        #pragma unroll
        for (int k = 0; k < TOPK; ++k) {          // iterative argmax; ties -> lowest index
            float best = -1.f; int bi = 0;
            for (int i = 0; i < EE; ++i) { float v = row[i]; if (v > best) { best = v; bi = i; } }
            ptop[k] = best * inv; itop[k] = bi;
            row[bi] = -1.f;                        // exclude (exp values are > 0)
        }
        float cs = 0.f; int ks = 0;
        #pragma unroll
        for (int k = 0; k < TOPK; ++k) { cs += ptop[k]; if (cs < TAUV) ks++; }
        if (ks < 1) ks = 1;
        float ssum = 0.f;
        #pragma unroll
        for (int k = 0; k < TOPK; ++k) if (k < ks) ssum += ptop[k];
        const float denom = fmaxf(ssum, 1e-7f);

        const size_t base = (size_t)(t0 + lane) * TOPK;
        #pragma unroll
        for (int k = 0; k < TOPK; ++k) {
            const bool keep = (k < ks);
            sel_idx[base + k] = keep ? itop[k] : -1;
            sel_p[base + k]   = keep ? ptop[k] / denom : 0.f;
        }
    }
}

// ---------------------------------------------------------------------------
// Kernel 2: zero-fill the two big output tensors. This is the roofline term
// (~335 MB of stores); use non-temporal (TH=NT) b128 stores so the dead
// zero lines stream to HBM instead of churning the 192 MB L2.
// ---------------------------------------------------------------------------
__global__ void fill_zero_kernel(v4f* __restrict__ out, size_t n4)
{
    size_t i = (size_t)blockIdx.x * blockDim.x + threadIdx.x;
    const size_t stride = (size_t)gridDim.x * blockDim.x;
    const v4f z = {0.f, 0.f, 0.f, 0.f};
    for (; i < n4; i += stride)
        __builtin_nontemporal_store(z, &out[i]);
}

// ---------------------------------------------------------------------------
// Kernel 3: globally ordered per-expert capacity scan + scatter + loss.
// Single 1024-thread block walks 131072 slots in order (128 chunks).
// ---------------------------------------------------------------------------
__global__ __launch_bounds__(1024)
void scatter_kernel(const int* __restrict__ sel_idx, const float* __restrict__ sel_p,
                    float* __restrict__ out)
{
    __shared__ int whist[32 * 64];   // per-wave per-expert counts within chunk
    __shared__ int wbase[32 * 64];   // exclusive base per wave per expert
    __shared__ int gcount[64];       // persistent global per-expert counters

    const int tid  = threadIdx.x;
    const int lane = tid & 31;
    const int wv   = tid >> 5;

    if (tid < 64) gcount[tid] = 0;
    __syncthreads();

    for (int chunk = 0; chunk < NSLOT / 1024; ++chunk) {
        const int slot = chunk * 1024 + tid;
        const int   e = sel_idx[slot];
        const float p = sel_p[slot];
        const bool valid = (e >= 0);

        whist[tid] = 0; whist[tid + 1024] = 0;
        __syncthreads();

        // ordered rank within the wave (count of lower lanes with same expert)
        int rank = 0;
        for (int j = 0; j < 32; ++j) {
            const int ej = __shfl(e, j, 32);
            if (j < lane && ej == e) rank++;
        }
        if (valid) atomicAdd(&whist[wv * 64 + e], 1);
        __syncthreads();

        // 64 threads: exclusive scan across the 32 waves for one expert each,
        // carrying the persistent global counter.
        if (tid < 64) {
            int run = gcount[tid];
            for (int w2 = 0; w2 < 32; ++w2) {
                wbase[w2 * 64 + tid] = run;
                run += whist[w2 * 64 + tid];
            }
            gcount[tid] = run;
        }
        __syncthreads();

        if (valid) {
            const int pos = wbase[wv * 64 + e] + rank;
            if (pos < CAP) {
                const int token = slot >> 3;                       // slot / TOPK
                const size_t off = ((size_t)token * EE + e) * CAP + pos;
                out[off]       = 1.0f;   // dispatch
                out[DSZ + off] = p;      // combine
            }
        }
        __syncthreads();
    }

    // load-balance loss from final counters: usage_e = min(count_e, CAP)
    if (tid == 0) {
        float s = 0.f; float u[64];
        #pragma unroll
        for (int i = 0; i < 64; ++i) {
            const int c = gcount[i];
            u[i] = (float)(c < CAP ? c : CAP);
            s += u[i];
        }
        const float mean = s / 64.f;
        float v = 0.f;
        #pragma unroll
        for (int i = 0; i < 64; ++i) { const float d = u[i] - mean; v += d * d; }
        v /= 64.f;
        out[2 * DSZ] = (s > 0.f) ? v / (mean + 1e-8f) : 0.f;
    }
}

// ---------------------------------------------------------------------------
extern "C" void kernel_launch(void* const* d_in, const int* in_sizes, int n_in,
                              void* d_out, int out_size, void* d_ws, size_t ws_size,
                              hipStream_t stream) {
    (void)in_sizes; (void)n_in; (void)out_size; (void)ws_size;
    const float* x = (const float*)d_in[0];     // [8,2048,1024] f32
    const float* w = (const float*)d_in[1];     // [1024,64] f32
    float* out = (float*)d_out;                 // dispatch | combine | loss

    int*   sel_idx = (int*)d_ws;                                          // 131072 ints
    float* sel_p   = (float*)((char*)d_ws + (size_t)NSLOT * sizeof(int)); // 131072 floats

    router_kernel<<<NTOK / 16, 32, 0, stream>>>(x, w, sel_idx, sel_p);
    fill_zero_kernel<<<8192, 256, 0, stream>>>((v4f*)out, (2 * DSZ) / 4);
    scatter_kernel<<<1, 1024, 0, stream>>>(sel_idx, sel_p, out);
}